// SELayer_14491219656853
// MI455X (gfx1250) — compile-verified
//
#include <hip/hip_runtime.h>
#include <hip/hip_bf16.h>

// SE layer: x(32,256,56,56) fp32
//   s    = mean(x, axis=(2,3))                  (32,256)
//   h    = relu(s @ w1.T + b1)                  (32,16)
//   gate = sigmoid(h @ w2.T + b2)               (32,256)
//   out  = x * gate[:, :, None, None]
//
// HBM-bound: ~309MB naive traffic; x fits in 192MB L2 so pass 3 re-reads from
// L2 and we stream the output with NT stores. The tiny MLP runs in fp32 on
// V_WMMA_F32_16X16X4_F32 in a single 2-wave workgroup.

#define SE_B     32
#define SE_C     256
#define SE_HID   16
#define SE_HW    3136          // 56*56
#define SE_HW4   784           // SE_HW / 4
#define SE_PLANES (SE_B * SE_C)

typedef __attribute__((ext_vector_type(2))) float v2f;
typedef __attribute__((ext_vector_type(4))) float v4f;
typedef __attribute__((ext_vector_type(8))) float v8f;

// ---------------------------------------------------------------------------
// Kernel 1: global average pool. One 256-thread block per (b,c) plane.
// ---------------------------------------------------------------------------
__global__ void __launch_bounds__(256) se_pool(const float* __restrict__ x,
                                               float* __restrict__ s) {
    const int plane = blockIdx.x;                       // b*256 + c
    const v4f* p4 = (const v4f*)(x + (size_t)plane * SE_HW);

    float sum = 0.0f;
    for (int i = threadIdx.x; i < SE_HW4; i += 256) {
        v4f v = p4[i];
        sum += (v.x + v.y) + (v.z + v.w);
    }

    __shared__ float red[256];
    red[threadIdx.x] = sum;
    __syncthreads();
    for (int off = 128; off >= 1; off >>= 1) {
        if (threadIdx.x < off) red[threadIdx.x] += red[threadIdx.x + off];
        __syncthreads();
    }
    if (threadIdx.x == 0) s[plane] = red[0] * (1.0f / (float)SE_HW);
}

// ---------------------------------------------------------------------------
// Kernel 2: full MLP on one workgroup (2 waves of 32), fp32 WMMA 16x16x4.
//   A layout (16x4 f32): lane l -> M = l&15; VGPR {0,1} hold K = 2*(l>>4)+{0,1}
//   B layout (4x16 f32): lane l -> N = l&15; VGPR {0,1} hold K = 2*(l>>4)+{0,1}
//   C/D (16x16 f32):     lane l -> N = l&15; VGPR r -> M = r + 8*(l>>4)
// ---------------------------------------------------------------------------
__global__ void __launch_bounds__(64) se_fc(const float* __restrict__ s,
                                            const float* __restrict__ w1,
                                            const float* __restrict__ b1,
                                            const float* __restrict__ w2,
                                            const float* __restrict__ b2,
                                            float* __restrict__ gate) {
    __shared__ float h_lds[SE_B][SE_HID];

    const int tid  = threadIdx.x;
    const int wave = tid >> 5;          // 0..1 -> batch tile
    const int lane = tid & 31;
    const int lm   = lane & 15;         // M (A) / N (B,C) index
    const int lh   = lane >> 4;         // lane half -> K offset 2*lh
    const int m0   = wave * 16;

    // ---- GEMM1: h(32x16) = S(32x256) @ W1^T(256x16), K = 256 ----
    // A[m][k] = s[m*256 + k];  B[k][n] = w1[n][k] = w1[n*256 + k]
    const float* arow = s  + (size_t)(m0 + lm) * SE_C + 2 * lh;
    const float* brow = w1 + (size_t)lm * SE_C + 2 * lh;

    v8f acc = {};
    for (int k0 = 0; k0 < SE_C; k0 += 4) {
        v2f a = *(const v2f*)(arow + k0);
        v2f b = *(const v2f*)(brow + k0);
        acc = __builtin_amdgcn_wmma_f32_16x16x4_f32(
            false, a, false, b, (short)0, acc, false, false);
    }

    // bias + relu, stage h tile to LDS for cross-lane relayout
    {
        const float bias1 = b1[lm];
#pragma unroll
        for (int r = 0; r < 8; ++r) {
            float hv = acc[r] + bias1;
            hv = hv > 0.0f ? hv : 0.0f;
            h_lds[m0 + r + 8 * lh][lm] = hv;
        }
    }
    __syncthreads();

    // ---- GEMM2: gate(32x256) = H(32x16) @ W2^T(16x256), K = 16 ----
    // A[m][k] = h_lds[m][k];  B[k][n] = w2[n][k] = w2[n*16 + k]
    for (int nt = 0; nt < 16; ++nt) {
        const int n0 = nt * 16;
        v8f c = {};
#pragma unroll
        for (int k0 = 0; k0 < SE_HID; k0 += 4) {
            v2f a;
            a.x = h_lds[m0 + lm][k0 + 2 * lh + 0];
            a.y = h_lds[m0 + lm][k0 + 2 * lh + 1];
            v2f b = *(const v2f*)(w2 + (size_t)(n0 + lm) * SE_HID + k0 + 2 * lh);
            c = __builtin_amdgcn_wmma_f32_16x16x4_f32(
                false, a, false, b, (short)0, c, false, false);
        }
        const float bias2 = b2[n0 + lm];
#pragma unroll
        for (int r = 0; r < 8; ++r) {
            float g = c[r] + bias2;
            g = 1.0f / (1.0f + expf(-g));
            const int m = m0 + r + 8 * lh;
            gate[(size_t)m * SE_C + (n0 + lm)] = g;
        }
    }
}

// ---------------------------------------------------------------------------
// Kernel 3: out = x * gate[plane]. x re-read hits L2 (pulled in by se_pool);
// output streamed with non-temporal stores to preserve x residency in L2.
// ---------------------------------------------------------------------------
__global__ void __launch_bounds__(256) se_scale(const float* __restrict__ x,
                                                const float* __restrict__ gate,
                                                float* __restrict__ out) {
    const int plane = blockIdx.x;                       // b*256 + c
    const float g = gate[plane];
    const v4f* p4 = (const v4f*)(x + (size_t)plane * SE_HW);
    v4f* o4 = (v4f*)(out + (size_t)plane * SE_HW);

    for (int i = threadIdx.x; i < SE_HW4; i += 256) {
        v4f v = p4[i];
        v *= g;
        __builtin_nontemporal_store(v, &o4[i]);
    }
}

// ---------------------------------------------------------------------------
// Launcher
// ---------------------------------------------------------------------------
extern "C" void kernel_launch(void* const* d_in, const int* in_sizes, int n_in,
                              void* d_out, int out_size, void* d_ws, size_t ws_size,
                              hipStream_t stream) {
    const float* x  = (const float*)d_in[0];   // (32,256,56,56)
    const float* w1 = (const float*)d_in[1];   // (16,256)
    const float* b1 = (const float*)d_in[2];   // (16)
    const float* w2 = (const float*)d_in[3];   // (256,16)
    const float* b2 = (const float*)d_in[4];   // (256)
    float* out = (float*)d_out;

    float* s_ws    = (float*)d_ws;             // 32*256 floats
    float* gate_ws = s_ws + SE_PLANES;         // 32*256 floats

    se_pool<<<SE_PLANES, 256, 0, stream>>>(x, s_ws);
    se_fc<<<1, 64, 0, stream>>>(s_ws, w1, b1, w2, b2, gate_ws);
    se_scale<<<SE_PLANES, 256, 0, stream>>>(x, gate_ws, out);
}